// YOLOv3_38190849196728
// MI455X (gfx1250) — compile-verified
//
#include <hip/hip_runtime.h>
#include <hip/hip_bf16.h>
#include <stdint.h>

// ---------------- problem constants (fixed by reference setup) ----------------
#define BATCH 8
#define NCLS 80
#define N0 (19*19*3)          // 1083
#define N1 (38*38*3)          // 4332
#define N2 (76*76*3)          // 17328
#define NTOT (N0+N1+N2)       // 22743
#define TOPK 100
#define SCORE_TH 0.5f
#define NMS_TH 0.3f
#define CAP2 4096             // per-class candidate cap (mean ~1.8k, >50 sigma margin)
#define NFLAT (NCLS*TOPK)     // 8000
#define SORT3 8192            // next pow2 >= 8000
#define NTHR 256

__device__ __forceinline__ float sigm(float x) { return 1.0f / (1.0f + expf(-x)); }

// CDNA5 async global->LDS 16B copy (per-lane), tracked by ASYNCcnt
__device__ __forceinline__ void async_copy_b128(unsigned lds_off, unsigned long long gaddr) {
    asm volatile("global_load_async_to_lds_b128 %0, %1, off"
                 :: "v"(lds_off), "v"(gaddr) : "memory");
}
__device__ __forceinline__ void async_wait0() {
    asm volatile("s_wait_asynccnt 0x0" ::: "memory");
}

// locate anchor n within the 3-layer pyramid
struct Loc {
    const float* f; size_t base; int cs; int gh; int a; int x; int y; int abase;
};
__device__ __forceinline__ Loc locate(int b, int n, const float* f0, const float* f1, const float* f2) {
    Loc L; int nl;
    if (n < N0)            { L.f = f0; L.gh = 19; L.abase = 6; nl = n; }
    else if (n < N0 + N1)  { L.f = f1; L.gh = 38; L.abase = 3; nl = n - N0; }
    else                   { L.f = f2; L.gh = 76; L.abase = 0; nl = n - (N0 + N1); }
    L.a = nl % 3; int cell = nl / 3;
    L.x = cell % L.gh; L.y = cell / L.gh;
    L.cs = L.gh * L.gh;
    L.base = ((size_t)b * 255 + (size_t)L.a * 85) * (size_t)L.cs + (size_t)L.y * L.gh + L.x;
    return L;
}

// LDS-resident bitonic sort, descending by u64 key
template<int SN>
__device__ __forceinline__ void bitonic_desc(unsigned long long* keys, int tid) {
    for (int k = 2; k <= SN; k <<= 1) {
        for (int j = k >> 1; j > 0; j >>= 1) {
            __syncthreads();
            for (int i = tid; i < SN; i += NTHR) {
                int l = i ^ j;
                if (l > i) {
                    unsigned long long a = keys[i], c = keys[l];
                    bool desc_blk = ((i & k) == 0);
                    bool sw = desc_blk ? (a < c) : (a > c);
                    if (sw) { keys[i] = c; keys[l] = a; }
                }
            }
        }
    }
    __syncthreads();
}

// ---------------- kernel 1: decode boxes (B*NTOT threads) ----------------
__global__ void yolo_decode_boxes(const float* __restrict__ f0, const float* __restrict__ f1,
                                  const float* __restrict__ f2, const float* __restrict__ anchors,
                                  const float* __restrict__ image_shape, float* __restrict__ boxes) {
    int t = blockIdx.x * blockDim.x + threadIdx.x;
    if (t >= BATCH * NTOT) return;
    int b = t / NTOT, n = t % NTOT;
    Loc L = locate(b, n, f0, f1, f2);
    const int cs = L.cs;
    float v0 = L.f[L.base + 0 * (size_t)cs];
    float v1 = L.f[L.base + 1 * (size_t)cs];
    float v2 = L.f[L.base + 2 * (size_t)cs];
    float v3 = L.f[L.base + 3 * (size_t)cs];

    const float in0 = 608.0f, in1 = 608.0f;      // feats0 19*32
    float im0 = image_shape[0], im1 = image_shape[1];

    float bx = (sigm(v0) + (float)L.x) / (float)L.gh;   // reference divides x by gh
    float by = (sigm(v1) + (float)L.y) / (float)L.gh;   // and y by gw (==gh here)
    float aw = anchors[(L.abase + L.a) * 2 + 0];
    float ah = anchors[(L.abase + L.a) * 2 + 1];
    float wdiv = expf(v2) * aw / in0;
    float hdiv = expf(v3) * ah / in1;

    // letterbox un-map (generic; degenerates to identity for 608x608)
    float r = fminf(in0 / im0, in1 / im1);
    float ns0 = rintf(im0 * r), ns1 = rintf(im1 * r);
    float off0 = (in0 - ns0) * 0.5f / in0, off1 = (in1 - ns1) * 0.5f / in1;
    float sc0 = in0 / ns0, sc1 = in1 / ns1;

    float yy = (by - off0) * sc0, xx = (bx - off1) * sc1;
    float hh = hdiv * sc0, ww = wdiv * sc1;
    float* o = boxes + (size_t)t * 4;
    o[0] = (yy - 0.5f * hh) * im0;
    o[1] = (xx - 0.5f * ww) * im1;
    o[2] = (yy + 0.5f * hh) * im0;
    o[3] = (xx + 0.5f * ww) * im1;
}

// -------- kernel 2: per-(image,class) threshold + top-100 + greedy NMS --------
__global__ __launch_bounds__(NTHR)
void yolo_topk_nms(const float* __restrict__ f0, const float* __restrict__ f1,
                   const float* __restrict__ f2, const float* __restrict__ boxes,
                   float* __restrict__ cand_boxes, float* __restrict__ flat_s) {
    const int tid = threadIdx.x;
    const int b = blockIdx.x / NCLS;
    const int c = blockIdx.x % NCLS;

    __shared__ unsigned long long keys[CAP2];
    __shared__ int cnt;
    __shared__ float sb_box[TOPK * 4];
    __shared__ float sval[TOPK];
    __shared__ int   skeep[TOPK];

    if (tid == 0) cnt = 0;
    for (int i = tid; i < CAP2; i += NTHR) keys[i] = 0ull;
    __syncthreads();

    // phase A: streaming threshold scan (conf/prob re-reads are L2-resident)
    for (int n = tid; n < NTOT; n += NTHR) {
        Loc L = locate(b, n, f0, f1, f2);
        const float* pconf = L.f + L.base + 4 * (size_t)L.cs;
        const float* pprob = L.f + L.base + (size_t)(5 + c) * L.cs;
        __builtin_prefetch(pconf + (size_t)NTHR, 0, 1);   // global_prefetch_b8
        float s = sigm(*pconf) * sigm(*pprob);
        if (s >= SCORE_TH) {
            int p = atomicAdd(&cnt, 1);
            if (p < CAP2)
                keys[p] = ((unsigned long long)__float_as_uint(s) << 32) |
                          (unsigned long long)(0xFFFFFFFFu - (unsigned)n);
        }
    }
    __syncthreads();

    // phase B: exact descending sort of candidates (ties -> lower index first)
    bitonic_desc<CAP2>(keys, tid);

    // phase C: extract top-100, async-gather their boxes straight into LDS
    if (tid < TOPK) {
        unsigned long long k = keys[tid];
        float v = __uint_as_float((unsigned)(k >> 32));
        int idx = (int)(0xFFFFFFFFu - (unsigned)(k & 0xFFFFFFFFull));
        int valid = (v >= SCORE_TH) ? 1 : 0;
        sval[tid] = v;
        skeep[tid] = valid;
        if (!valid) idx = 0;                                  // pad slot: harmless gather
        unsigned lds_off = (unsigned)(size_t)(&sb_box[tid * 4]);
        unsigned long long gaddr =
            (unsigned long long)(size_t)(boxes + ((size_t)b * NTOT + (size_t)idx) * 4);
        async_copy_b128(lds_off, gaddr);
    }
    async_wait0();
    __syncthreads();

    volatile float* vb = sb_box;   // written by async DMA: force real LDS reads

    // phase D: greedy NMS, exact reference fori_loop semantics
    for (int i = 0; i < TOPK - 1; i++) {
        __syncthreads();
        int live = skeep[i];
        float bi0 = vb[i * 4 + 0], bi1 = vb[i * 4 + 1];
        float bi2 = vb[i * 4 + 2], bi3 = vb[i * 4 + 3];
        if (live && tid > i && tid < TOPK && skeep[tid]) {
            float bj0 = vb[tid * 4 + 0], bj1 = vb[tid * 4 + 1];
            float bj2 = vb[tid * 4 + 2], bj3 = vb[tid * 4 + 3];
            float ai = fmaxf(bi2 - bi0, 0.0f) * fmaxf(bi3 - bi1, 0.0f);
            float aj = fmaxf(bj2 - bj0, 0.0f) * fmaxf(bj3 - bj1, 0.0f);
            float tlx = fmaxf(bi0, bj0), tly = fmaxf(bi1, bj1);
            float brx = fminf(bi2, bj2), bry = fminf(bi3, bj3);
            float inter = fmaxf(brx - tlx, 0.0f) * fmaxf(bry - tly, 0.0f);
            float iou = inter / (ai + aj - inter + 1e-9f);
            if (iou > NMS_TH) skeep[tid] = 0;
        }
    }
    __syncthreads();

    // phase E: emit candidates for the per-image merge
    if (tid < TOPK) {
        size_t base = (size_t)blockIdx.x * TOPK + tid;    // b*8000 + c*100 + slot
        cand_boxes[base * 4 + 0] = vb[tid * 4 + 0];
        cand_boxes[base * 4 + 1] = vb[tid * 4 + 1];
        cand_boxes[base * 4 + 2] = vb[tid * 4 + 2];
        cand_boxes[base * 4 + 3] = vb[tid * 4 + 3];
        flat_s[base] = skeep[tid] ? sval[tid] : -1.0f;
    }
}

// -------- kernel 3: per-image top-100 over 8000 + letterbox rescale --------
__global__ __launch_bounds__(NTHR)
void yolo_final(const float* __restrict__ cand_boxes, const float* __restrict__ flat_s,
                const float* __restrict__ image_shape, const float* __restrict__ ori_shape,
                float* __restrict__ out) {
    const int tid = threadIdx.x;
    const int b = blockIdx.x;
    __shared__ unsigned long long keys[SORT3];   // 64 KB
    __shared__ float stage[NFLAT];               // 32 KB, filled by async DMA

    // bulk async global->LDS stream of this image's masked scores (2000 x b128)
    const float* src = flat_s + (size_t)b * NFLAT;
    for (int chunk = tid; chunk < NFLAT / 4; chunk += NTHR) {
        unsigned lds_off = (unsigned)(size_t)(&stage[chunk * 4]);
        unsigned long long gaddr = (unsigned long long)(size_t)(src + (size_t)chunk * 4);
        async_copy_b128(lds_off, gaddr);
    }
    async_wait0();
    __syncthreads();

    volatile float* vs = stage;                  // DMA-written: force real LDS reads
    for (int i = tid; i < SORT3; i += NTHR) {
        unsigned long long k = 0ull;
        if (i < NFLAT) {
            float v = vs[i];
            unsigned bits = __float_as_uint(v);
            unsigned ord = (bits & 0x80000000u) ? ~bits : (bits | 0x80000000u);
            k = ((unsigned long long)ord << 32) |
                (unsigned long long)(0xFFFFFFFFu - (unsigned)i);
        }
        keys[i] = k;
    }
    __syncthreads();
    bitonic_desc<SORT3>(keys, tid);

    if (tid < TOPK) {
        unsigned long long k = keys[tid];
        unsigned ord = (unsigned)(k >> 32);
        unsigned bits = (ord & 0x80000000u) ? (ord ^ 0x80000000u) : ~ord;
        float v = __uint_as_float(bits);
        int flatidx = (int)(0xFFFFFFFFu - (unsigned)(k & 0xFFFFFFFFull));
        bool valid = (v >= SCORE_TH);            // pads decode to NaN -> false
        int gi = valid ? flatidx : 0;
        const float* dbx = cand_boxes + ((size_t)b * NFLAT + (size_t)gi) * 4;
        float d0 = dbx[0], d1 = dbx[1], d2 = dbx[2], d3 = dbx[3];

        float w = image_shape[0], h = image_shape[1];
        float iw = ori_shape[b * 2 + 0], ih = ori_shape[b * 2 + 1];
        float scale = fminf(w / iw, h / ih);
        float nw = floorf(iw * scale), nh = floorf(ih * scale);
        float dx = floorf((w - nw) * 0.5f), dy = floorf((h - nh) * 0.5f);

        float o0 = (d0 - dx) / scale, o1 = (d1 - dy) / scale;
        float o2 = (d2 - dx) / scale, o3 = (d3 - dy) / scale;
        int cls = flatidx / TOPK + 1;

        size_t r = (size_t)b * TOPK + tid;
        if (valid) {
            out[r * 5 + 0] = o0; out[r * 5 + 1] = o1;
            out[r * 5 + 2] = o2; out[r * 5 + 3] = o3;
            out[r * 5 + 4] = v;
            out[BATCH * TOPK * 5 + r] = (float)cls;
            out[BATCH * TOPK * 6 + r] = 1.0f;
        } else {
            out[r * 5 + 0] = 0.0f; out[r * 5 + 1] = 0.0f;
            out[r * 5 + 2] = 0.0f; out[r * 5 + 3] = 0.0f;
            out[r * 5 + 4] = 0.0f;
            out[BATCH * TOPK * 5 + r] = 0.0f;
            out[BATCH * TOPK * 6 + r] = 0.0f;
        }
    }
}

// ---------------- host launcher ----------------
extern "C" void kernel_launch(void* const* d_in, const int* in_sizes, int n_in,
                              void* d_out, int out_size, void* d_ws, size_t ws_size,
                              hipStream_t stream) {
    const float* f0 = (const float*)d_in[0];
    const float* f1 = (const float*)d_in[1];
    const float* f2 = (const float*)d_in[2];
    const float* anchors = (const float*)d_in[3];
    const float* image_shape = (const float*)d_in[4];
    const float* ori_shape = (const float*)d_in[5];
    float* out = (float*)d_out;

    float* boxes = (float*)d_ws;                                  // B*NTOT*4
    float* cand_boxes = boxes + (size_t)BATCH * NTOT * 4;         // B*80*100*4
    float* flat_s = cand_boxes + (size_t)BATCH * NFLAT * 4;       // B*80*100
    (void)in_sizes; (void)n_in; (void)out_size; (void)ws_size;

    int total = BATCH * NTOT;
    yolo_decode_boxes<<<(total + NTHR - 1) / NTHR, NTHR, 0, stream>>>(
        f0, f1, f2, anchors, image_shape, boxes);
    yolo_topk_nms<<<BATCH * NCLS, NTHR, 0, stream>>>(
        f0, f1, f2, boxes, cand_boxes, flat_s);
    yolo_final<<<BATCH, NTHR, 0, stream>>>(
        cand_boxes, flat_s, image_shape, ori_shape, out);
}